// GraphConv_75917841924563
// MI455X (gfx1250) — compile-verified
//
#include <hip/hip_runtime.h>
#include <math.h>

#define N_ENT    50000
#define C_DIM    128
#define N_EDGES  1600000
#define N_RELM1  10          // N_REL - 1
#define EPSV     1e-12f

#define WAVES_PER_BLOCK  8   // 256 threads = 8 wave32
#define EDGES_PER_WAVE   8
#define EDGES_PER_BLOCK  (WAVES_PER_BLOCK * EDGES_PER_WAVE)   // 64

// Flip to 0 if the gfx1250 async mnemonic is rejected by the assembler.
#define USE_ASYNC_LDS 1

// -------------------- utility kernels --------------------

__global__ void zero_f4_kernel(float4* __restrict__ p, int n4) {
    int i = blockIdx.x * blockDim.x + threadIdx.x;
    if (i < n4) p[i] = make_float4(0.f, 0.f, 0.f, 0.f);
}

__global__ void copy_f4_kernel(const float4* __restrict__ s, float4* __restrict__ d, int n4) {
    int i = blockIdx.x * blockDim.x + threadIdx.x;
    if (i < n4) d[i] = s[i];
}

// per-head in-degree (float: exact for counts < 2^24)
__global__ void count_heads_kernel(const int* __restrict__ head, float* __restrict__ cnt) {
    int e = blockIdx.x * blockDim.x + threadIdx.x;
    if (e < N_EDGES) {
        unsafeAtomicAdd(&cnt[head[e]], 1.0f);   // global_atomic_add_f32, no return
    }
}

// -------------------- scatter: agg[head] += emb[tail] * W[etype-1] --------------------

__global__ void __launch_bounds__(256)
scatter_kernel(const float* __restrict__ emb,
               const float* __restrict__ weight,
               const int*   __restrict__ head,
               const int*   __restrict__ tail,
               const int*   __restrict__ etype,
               float*       __restrict__ agg) {
    // Sole __shared__ object in this kernel -> LDS byte offset 0.
    __shared__ float wlds[N_RELM1 * C_DIM];      // 5 KB relation table in LDS

#if USE_ASYNC_LDS
    // Direct memory->LDS async copy (gfx1250): 5120 B = 320 lanes x B128.
    // vdst = LDS byte address (array base offset 0), vaddr = same byte offset
    // from the uniform SGPR base `weight`. Tracked by ASYNCcnt.
    for (int idx = threadIdx.x; idx < (N_RELM1 * C_DIM) / 4; idx += 256) {
        unsigned off = (unsigned)idx * 16u;
        asm volatile("global_load_async_to_lds_b128 %0, %1, %2"
                     :
                     : "v"(off), "v"(off), "s"(weight)
                     : "memory");
    }
    asm volatile("s_wait_asynccnt 0x0" ::: "memory");
#else
    for (int i = threadIdx.x; i < N_RELM1 * C_DIM; i += blockDim.x)
        wlds[i] = weight[i];
#endif
    __syncthreads();

    const int wave = threadIdx.x >> 5;
    const int lane = threadIdx.x & 31;
    const int c4   = lane << 2;                  // 4 channels per lane -> 128/wave
    const int ebase = blockIdx.x * EDGES_PER_BLOCK + wave * EDGES_PER_WAVE;

    for (int i = 0; i < EDGES_PER_WAVE; ++i) {
        int e = ebase + i;
        if (e >= N_EDGES) return;

        // stream-ahead prefetch of edge lists (speculative -> OOB safe)
        __builtin_prefetch(&head[e + 4096], 0, 0);
        __builtin_prefetch(&tail[e + 4096], 0, 0);
        __builtin_prefetch(&etype[e + 4096], 0, 0);

        const int h = head[e];
        const int t = tail[e];
        const int r = etype[e] - 1;

        const float4 v = *(const float4*)(emb  + t * C_DIM + c4);   // global_load_b128 (L2-resident)
        const float4 w = *(const float4*)(wlds + r * C_DIM + c4);   // ds_load_b128

        float* dst = agg + h * C_DIM + c4;
        unsafeAtomicAdd(dst + 0, v.x * w.x);     // global_atomic_add_f32 @ L2
        unsafeAtomicAdd(dst + 1, v.y * w.y);
        unsafeAtomicAdd(dst + 2, v.z * w.z);
        unsafeAtomicAdd(dst + 3, v.w * w.w);
    }
}

// -------------------- fused mean + L2 normalize (in place) + res accumulate --------------------

__global__ void __launch_bounds__(256)
normalize_add_kernel(float* __restrict__ buf,         // agg in, normalized emb out
                     const float* __restrict__ cnt,
                     float* __restrict__ res) {
    const int wave = threadIdx.x >> 5;
    const int lane = threadIdx.x & 31;
    const int row  = blockIdx.x * WAVES_PER_BLOCK + wave;
    if (row >= N_ENT) return;

    float4* p = (float4*)(buf + row * C_DIM) + lane;
    float4 v = *p;

    const float inv = 1.0f / fmaxf(cnt[row], 1.0f);   // scatter_mean denom
    v.x *= inv; v.y *= inv; v.z *= inv; v.w *= inv;

    // row sum of squares: 4 local + 5-step wave32 butterfly
    float ss = v.x * v.x + v.y * v.y + v.z * v.z + v.w * v.w;
    #pragma unroll
    for (int off = 16; off > 0; off >>= 1)
        ss += __shfl_xor(ss, off, 32);

    const float scale = 1.0f / fmaxf(sqrtf(ss), EPSV);   // zero rows stay zero
    v.x *= scale; v.y *= scale; v.z *= scale; v.w *= scale;
    *p = v;

    float4* rp = (float4*)(res + row * C_DIM) + lane;
    float4 rv = *rp;
    rv.x += v.x; rv.y += v.y; rv.z += v.z; rv.w += v.w;
    *rp = rv;
}

// -------------------- host driver --------------------

extern "C" void kernel_launch(void* const* d_in, const int* in_sizes, int n_in,
                              void* d_out, int out_size, void* d_ws, size_t ws_size,
                              hipStream_t stream) {
    const float* entity_emb = (const float*)d_in[0];
    const int*   edge_index = (const int*)d_in[1];   // [2, E] flat
    const int*   edge_type  = (const int*)d_in[2];
    const float* weight     = (const float*)d_in[3];

    const int* head = edge_index;
    const int* tail = edge_index + N_EDGES;
    float*     res  = (float*)d_out;

    // workspace layout (all 16B aligned: 200000 % 16 == 0)
    float* cnt  = (float*)d_ws;                       // N_ENT
    float* bufA = cnt + N_ENT;                        // N_ENT * C
    float* bufB = bufA + (size_t)N_ENT * C_DIM;       // N_ENT * C

    const int n4_emb = N_ENT * C_DIM / 4;             // 1,600,000
    const int n4_cnt = N_ENT / 4;                     // 12,500
    const int edgeGrid  = (N_EDGES + 255) / 256;                                 // 6250
    const int scatGrid  = (N_EDGES + EDGES_PER_BLOCK - 1) / EDGES_PER_BLOCK;     // 25000
    const int normGrid  = (N_ENT + WAVES_PER_BLOCK - 1) / WAVES_PER_BLOCK;       // 6250
    const int embGrid   = (n4_emb + 255) / 256;                                  // 6250

    // in-degree counts (constant across hops)
    zero_f4_kernel<<<(n4_cnt + 255) / 256, 256, 0, stream>>>((float4*)cnt, n4_cnt);
    count_heads_kernel<<<edgeGrid, 256, 0, stream>>>(head, cnt);

    // res = entity_emb
    copy_f4_kernel<<<embGrid, 256, 0, stream>>>((const float4*)entity_emb, (float4*)res, n4_emb);

    // hop 1: entity_emb -> bufA
    zero_f4_kernel<<<embGrid, 256, 0, stream>>>((float4*)bufA, n4_emb);
    scatter_kernel<<<scatGrid, 256, 0, stream>>>(entity_emb, weight, head, tail, edge_type, bufA);
    normalize_add_kernel<<<normGrid, 256, 0, stream>>>(bufA, cnt, res);

    // hop 2: bufA -> bufB
    zero_f4_kernel<<<embGrid, 256, 0, stream>>>((float4*)bufB, n4_emb);
    scatter_kernel<<<scatGrid, 256, 0, stream>>>(bufA, weight, head, tail, edge_type, bufB);
    normalize_add_kernel<<<normGrid, 256, 0, stream>>>(bufB, cnt, res);

    // hop 3: bufB -> bufA
    zero_f4_kernel<<<embGrid, 256, 0, stream>>>((float4*)bufA, n4_emb);
    scatter_kernel<<<scatGrid, 256, 0, stream>>>(bufB, weight, head, tail, edge_type, bufA);
    normalize_add_kernel<<<normGrid, 256, 0, stream>>>(bufA, cnt, res);
}